// ChamferLoss_60344290509660
// MI455X (gfx1250) — compile-verified
//
#include <hip/hip_runtime.h>

typedef __attribute__((ext_vector_type(2))) float v2f;
typedef __attribute__((ext_vector_type(8))) float v8f;

#define N_PTS 4096
#define M_PTS 4096
#define TILES_M (M_PTS / 16)
#define STRIPS 256                        // N_PTS / 16 row-strips per batch
#define STRIPS_PER_WAVE 2
#define WAVES_PER_BLOCK 8
#define STRIP_PAIRS (STRIPS / STRIPS_PER_WAVE)              // 128 wave-jobs per batch
#define BLOCKS_PER_BATCH (STRIP_PAIRS / WAVES_PER_BLOCK)    // 16

__global__ void ChamferZeroOut(float* out) {
    out[0] = 0.0f;
}

__global__ void __launch_bounds__(256)
ChamferWmmaKernel(const float* __restrict__ x,
                  const float* __restrict__ y,
                  float* __restrict__ out) {
    const int lane  = threadIdx.x & 31;
    const int wave  = threadIdx.x >> 5;
    const int batch = blockIdx.x / BLOCKS_PER_BATCH;
    const int pair  = (blockIdx.x % BLOCKS_PER_BATCH) * WAVES_PER_BLOCK + wave; // 0..127
    const int strip0 = pair * STRIPS_PER_WAVE;       // first 16-row strip
    const int strip1 = strip0 + 1;                   // second 16-row strip
    const int r     = lane & 15;                     // row within 16-tile, both halves
    const bool hi   = lane >= 16;

    // ---- Build A' (16x4 f32, 2 VGPRs each) for both strips ----
    // A'[m][k] = -2*x[m][k] for k<3, A'[m][3] = 1
    const float* px0 = x + ((size_t)batch * N_PTS + (size_t)strip0 * 16 + r) * 3;
    const float* px1 = x + ((size_t)batch * N_PTS + (size_t)strip1 * 16 + r) * 3;
    const float x00 = px0[0], x01 = px0[1], x02 = px0[2];
    const float x10 = px1[0], x11 = px1[1], x12 = px1[2];
    const float xnorm0 = x00 * x00 + x01 * x01 + x02 * x02;
    const float xnorm1 = x10 * x10 + x11 * x11 + x12 * x12;

    v2f a0, a1;
    a0.x = hi ? (-2.0f * x02) : (-2.0f * x00);   // VGPR0: K=0 (lo) / K=2 (hi)
    a0.y = hi ? 1.0f          : (-2.0f * x01);   // VGPR1: K=1 (lo) / K=3 (hi)
    a1.x = hi ? (-2.0f * x12) : (-2.0f * x10);
    a1.y = hi ? 1.0f          : (-2.0f * x11);

    float runmin0[8], runmin1[8];
#pragma unroll
    for (int i = 0; i < 8; ++i) { runmin0[i] = __builtin_inff(); runmin1[i] = __builtin_inff(); }

    const v8f cz = {};  // zero accumulator, reused every tile

    // ---- Sweep all 256 y-tiles of this batch; each B tile feeds 2 WMMAs ----
    const float* py = y + ((size_t)batch * M_PTS + r) * 3;
    for (int t = 0; t < TILES_M; ++t) {
        const float y0 = py[0], y1 = py[1], y2 = py[2];
        py += 16 * 3;
        const float ynorm = y0 * y0 + y1 * y1 + y2 * y2;

        // B' (4x16 f32, 2 VGPRs): B'[k][n] = y[n][k] (k<3), B'[3][n] = ||y[n]||^2
        v2f b;
        b.x = hi ? y2    : y0;   // VGPR0: K=0 (lo) / K=2 (hi)
        b.y = hi ? ynorm : y1;   // VGPR1: K=1 (lo) / K=3 (hi)

        // D[m][n] = -2*x.y + ||y[n]||^2
        v8f d0 = __builtin_amdgcn_wmma_f32_16x16x4_f32(
            false, a0, false, b, (short)0, cz, false, false);
        v8f d1 = __builtin_amdgcn_wmma_f32_16x16x4_f32(
            false, a1, false, b, (short)0, cz, false, false);

#pragma unroll
        for (int i = 0; i < 8; ++i) {
            runmin0[i] = fminf(runmin0[i], d0[i]);
            runmin1[i] = fminf(runmin1[i], d1[i]);
        }
    }

    // ---- Column-min reduction within each 16-lane half, then add ||x||^2 ----
    // C/D layout: VGPR v -> row m = v (lanes 0-15) or v+8 (lanes 16-31), n = lane%16.
    float acc = 0.0f;
#pragma unroll
    for (int v = 0; v < 8; ++v) {
        float rm0 = runmin0[v];
        float rm1 = runmin1[v];
        rm0 = fminf(rm0, __shfl_xor(rm0, 1, 32));
        rm1 = fminf(rm1, __shfl_xor(rm1, 1, 32));
        rm0 = fminf(rm0, __shfl_xor(rm0, 2, 32));
        rm1 = fminf(rm1, __shfl_xor(rm1, 2, 32));
        rm0 = fminf(rm0, __shfl_xor(rm0, 4, 32));
        rm1 = fminf(rm1, __shfl_xor(rm1, 4, 32));
        rm0 = fminf(rm0, __shfl_xor(rm0, 8, 32));
        rm1 = fminf(rm1, __shfl_xor(rm1, 8, 32));  // all 16 lanes of each half hold row min
        const int row = v + (hi ? 8 : 0);          // 0..15
        const float xn0 = __shfl(xnorm0, row, 32); // lanes 0..15 hold norms of rows 0..15
        const float xn1 = __shfl(xnorm1, row, 32);
        acc += (rm0 + xn0) + (rm1 + xn1);
    }

    // Lower half's acc = rows v=0..7 of both strips, upper half's = rows 8..15.
    const float total = __shfl(acc, 0, 32) + __shfl(acc, 16, 32);
    if (lane == 0) {
        atomicAdd(out, total * (1.0f / (float)N_PTS));
    }
}

extern "C" void kernel_launch(void* const* d_in, const int* in_sizes, int n_in,
                              void* d_out, int out_size, void* d_ws, size_t ws_size,
                              hipStream_t stream) {
    const float* x = (const float*)d_in[0];
    const float* y = (const float*)d_in[1];
    float* out = (float*)d_out;

    ChamferZeroOut<<<1, 1, 0, stream>>>(out);

    const int blocks = 16 * BLOCKS_PER_BATCH;  // 256
    ChamferWmmaKernel<<<blocks, 256, 0, stream>>>(x, y, out);
}